// FMModel_9053791060316
// MI455X (gfx1250) — compile-verified
//
#include <hip/hip_runtime.h>
#include <math.h>
#include <stdint.h>

// Problem constants (from reference)
#define NF 24
#define VOC 100000
#define DIM 64
#define BATCH 16384

#define WPB 8                      // waves per block
#define TPB (WPB * 32)             // 256 threads
#define ROWB (DIM * 4)             // 256 bytes per embedding row
#define BUF (NF * DIM)             // 1536 floats per sample staging buffer
#define NBLK 512                   // 4096 waves -> 4 samples per wave (pipelined)

__global__ __launch_bounds__(TPB) void fm_fused_kernel(
    const int*   __restrict__ x,      // [B, NF]
    const float* __restrict__ Wemb,   // [NF, VOC, DIM]
    const float* __restrict__ Wlin,   // [NF, VOC]
    const float* __restrict__ bias,   // [1]
    float*       __restrict__ out)    // [B, 1]
{
    __shared__ __align__(16) float smem[WPB * 2 * BUF];   // 96 KB: double buffer per wave

    const int lane = threadIdx.x & 31;
    const int wv   = threadIdx.x >> 5;
    const int gw   = blockIdx.x * WPB + wv;     // global wave id (wave-uniform)
    const int nw   = gridDim.x * WPB;

    const int c16  = (lane & 15) << 4;          // this lane's 16B chunk within a row
    const int fhal = lane >> 4;                 // half-wave selects even/odd field

    const float bz = bias[0];

    int b = gw;
    if (b >= BATCH) return;                     // wave-uniform

    // ---- prologue: async-gather sample b's 24 rows into buffer 0 ----
    int xcur = (lane < NF) ? x[b * NF + lane] : 0;
    {
        uint32_t lbase = (uint32_t)(uintptr_t)&smem[(wv * 2 + 0) * BUF];
#pragma unroll
        for (int j = 0; j < 12; ++j) {
            int f   = 2 * j + fhal;                         // field for this half-wave
            int idx = __shfl(xcur, f, 32);                  // broadcast x[b,f]
            uint64_t g = (uint64_t)(uintptr_t)Wemb
                       + ((uint64_t)f * VOC + (uint64_t)idx) * ROWB + c16;
            uint32_t l = lbase + (uint32_t)(f * ROWB) + (uint32_t)c16;
            asm volatile("global_load_async_to_lds_b128 %0, %1, off"
                         :: "v"(l), "v"(g) : "memory");
        }
    }

    int p = 0;
    for (; b < BATCH; b += nw) {
        const int bn = b + nw;
        int xnext = 0;

        // ---- kick async gathers for the NEXT sample into the other buffer ----
        if (bn < BATCH) {
            xnext = (lane < NF) ? x[bn * NF + lane] : 0;
            uint32_t lbase = (uint32_t)(uintptr_t)&smem[(wv * 2 + (p ^ 1)) * BUF];
#pragma unroll
            for (int j = 0; j < 12; ++j) {
                int f   = 2 * j + fhal;
                int idx = __shfl(xnext, f, 32);
                uint64_t g = (uint64_t)(uintptr_t)Wemb
                           + ((uint64_t)f * VOC + (uint64_t)idx) * ROWB + c16;
                uint32_t l = lbase + (uint32_t)(f * ROWB) + (uint32_t)c16;
                asm volatile("global_load_async_to_lds_b128 %0, %1, off"
                             :: "v"(l), "v"(g) : "memory");
            }
        }

        // first-order gather for the CURRENT sample (overlaps with async wait)
        float wl = (lane < NF) ? Wlin[(size_t)lane * VOC + (size_t)xcur] : 0.0f;

        // current sample's 12 async loads retire first (in-order); next 12 may remain
        if (bn < BATCH) {
            asm volatile("s_wait_asynccnt 0xc" ::: "memory");
        } else {
            asm volatile("s_wait_asynccnt 0x0" ::: "memory");
        }

        // ---- reduce current sample from LDS: lane owns dims 2*lane, 2*lane+1 ----
        const float* bp = &smem[(wv * 2 + p) * BUF];
        float a0 = 0.0f, a1 = 0.0f, ssq = 0.0f;
#pragma unroll
        for (int f = 0; f < NF; ++f) {
            const float2 e = *(const float2*)&bp[f * DIM + 2 * lane];   // ds_load_b64, bank-conflict-free
            a0 += e.x;
            a1 += e.y;
            ssq = fmaf(e.x, e.x, ssq);
            ssq = fmaf(e.y, e.y, ssq);
        }

        // per-lane partial of  0.5*(||sum||^2 - sum||e||^2) + first_order
        float part = 0.5f * (a0 * a0 + a1 * a1 - ssq) + wl;
#pragma unroll
        for (int off = 16; off; off >>= 1)
            part += __shfl_xor(part, off, 32);

        if (lane == 0)
            out[b] = 1.0f / (1.0f + __expf(-(bz + part)));

        xcur = xnext;
        p ^= 1;
    }
}

extern "C" void kernel_launch(void* const* d_in, const int* in_sizes, int n_in,
                              void* d_out, int out_size, void* d_ws, size_t ws_size,
                              hipStream_t stream) {
    const int*   x    = (const int*)d_in[0];
    const float* Wemb = (const float*)d_in[1];
    const float* Wlin = (const float*)d_in[2];
    const float* bias = (const float*)d_in[3];
    float*       out  = (float*)d_out;
    fm_fused_kernel<<<NBLK, TPB, 0, stream>>>(x, Wemb, Wlin, bias, out);
}